// DistanceTransform_34007551050116
// MI455X (gfx1250) — compile-verified
//
#include <hip/hip_runtime.h>
#include <stdint.h>
#include <math.h>

// ---------------------------------------------------------------------------
// Continuous distance transform (softmin wavefront), MI455X / gfx1250.
//
// Reduction: boundary is binary forever; each pixel activates exactly once,
// the first iteration it has >=1 active 8-neighbor, with
//     out = iter + lut[ne][nc],   lut[ne][nc] = -h*log(ne*E + nc*C)
// (E = exp(-1/h) edge weight, C = exp(-sqrt2/h) corner weight, ne/nc in 0..4).
// State = 256x256 bits/image -> double-buffered 64-bit bitboards in LDS.
// Ingest uses the Tensor Data Mover (tensor_load_to_lds, TENSORcnt),
// double-buffered so DMA overlaps the float->bit conversion.
// Early-exit: once no word changes, all remaining reference iterations are
// provable no-ops, so breaking is exact and input-deterministic.
// ---------------------------------------------------------------------------

#define H_IMG    256
#define W_IMG    256
#define W64PR    4                   // 64-bit words per row
#define NW64     (H_IMG * W64PR)     // 1024 u64 words per image
#define NTHREADS 1024                // one u64 word per thread
#define N_ITERS  256                 // ceil(max(H,W)/kh), kh=1
#define HPARAM   0.35f

typedef unsigned int u32x4 __attribute__((ext_vector_type(4)));
typedef int          i32x4 __attribute__((ext_vector_type(4)));
typedef int          i32x8 __attribute__((ext_vector_type(8)));

// bit j of result = value at column (c-1); replicate padding at image edge
__device__ __forceinline__ uint64_t west64(uint64_t x, uint64_t left, int wc) {
    uint64_t carry = (wc == 0) ? (x & 1ull) : (left >> 63);
    return (x << 1) | carry;
}
// bit j of result = value at column (c+1); replicate padding at image edge
__device__ __forceinline__ uint64_t east64(uint64_t x, uint64_t right, int wc) {
    uint64_t carry = (wc == W64PR - 1) ? (x >> 63) : (right & 1ull);
    return (x >> 1) | (carry << 63);
}
// bit-sliced sum of four 1-bit planes -> 3-bit count planes (0..4 per bit)
__device__ __forceinline__ void popplanes4(uint64_t a, uint64_t b, uint64_t c, uint64_t d,
                                           uint64_t &p0, uint64_t &p1, uint64_t &p2) {
    uint64_t s1 = a ^ b, c1 = a & b;
    uint64_t s2 = c ^ d, c2 = c & d;
    p0 = s1 ^ s2;
    uint64_t cr = s1 & s2;
    p1 = c1 ^ c2 ^ cr;
    p2 = (c1 & c2) | (cr & (c1 ^ c2));
}

// Issue one TDM load of an 8-row f32 slab (8 KB) into LDS.
__device__ __forceinline__ void tdm_load_slab(const float* gsrc, uint32_t ldsAddr) {
    unsigned long long ga = (unsigned long long)(uintptr_t)gsrc;
    u32x4 g0;
    g0[0] = 1u;                                     // count=1
    g0[1] = ldsAddr;                                // lds_addr
    g0[2] = (uint32_t)ga;                           // global_addr[31:0]
    g0[3] = ((uint32_t)(ga >> 32) & 0x01FFFFFFu)    // global_addr[56:32]
            | 0x80000000u;                          // type=2 ("image")
    i32x8 g1;
    g1[0] = 0x00020000;                             // data_size = 4 bytes
    g1[1] = (int)(W_IMG << 16);                     // tensor_dim0 = 256
    g1[2] = (int)(H_IMG << 16);                     // tensor_dim1 = 256
    g1[3] = (int)(W_IMG << 16);                     // tile_dim0   = 256
    g1[4] = 8;                                      // tile_dim1   = 8 rows
    g1[5] = W_IMG;                                  // tensor_dim0_stride = 256
    g1[6] = 0;
    g1[7] = 0;
    i32x4 z4 = {0, 0, 0, 0};
#if defined(__clang_major__) && (__clang_major__ >= 23)
    i32x8 z8 = {0, 0, 0, 0, 0, 0, 0, 0};
    __builtin_amdgcn_tensor_load_to_lds(g0, g1, z4, z4, z8, 0);
#else
    __builtin_amdgcn_tensor_load_to_lds(g0, g1, z4, z4, 0);
#endif
}

__global__ __launch_bounds__(NTHREADS)
void cdt_wavefront_kernel(const float* __restrict__ img, float* __restrict__ out) {
    __shared__ uint64_t Bb[2][NW64];          // double-buffered bitboard (16 KB)
    __shared__ float    stage[2][8 * W_IMG];  // double-buffered TDM staging (16 KB)
    __shared__ float    lut[25];
    __shared__ int      chg[2];               // parity-paired "changed" flags

    const int tid = threadIdx.x;
    const int b   = blockIdx.x;
    const float* im = img + (size_t)b * (H_IMG * W_IMG);
    float*       ob = out + (size_t)b * (H_IMG * W_IMG);

    if (tid < 25) {
        const float ew = expf(-1.0f / HPARAM);
        const float cw = expf(-sqrtf(2.0f) / HPARAM);
        int ne = tid / 5, nc = tid % 5;
        float sv = (float)ne * ew + (float)nc * cw;
        lut[tid] = (sv > 0.0f) ? (-HPARAM * logf(sv)) : 0.0f;
    }

    // Zero the output with b128 stores (seeds & unreached pixels stay 0);
    // fence so these complete before any activation store to the same bytes.
    float4* ob4 = (float4*)ob;
    float4  z4f = make_float4(0.f, 0.f, 0.f, 0.f);
    for (int k = tid; k < (H_IMG * W_IMG) / 4; k += NTHREADS) ob4[k] = z4f;
    __threadfence();

    // ---- ingest: pipelined TDM loads of 8-row slabs, convert to bitboard ----
    const uint32_t ldsStage0 = (uint32_t)(uintptr_t)(void*)&stage[0][0];
    const uint32_t ldsStage1 = (uint32_t)(uintptr_t)(void*)&stage[1][0];
    if (tid < 32) tdm_load_slab(im, ldsStage0);            // prologue: slab 0
    for (int blk = 0; blk < H_IMG / 8; ++blk) {
        if (tid < 32) {                                    // wave 0, EXEC all-1s
            if (blk + 1 < H_IMG / 8) {
                tdm_load_slab(im + (size_t)(blk + 1) * 8 * W_IMG,
                              ((blk + 1) & 1) ? ldsStage1 : ldsStage0);
                __builtin_amdgcn_s_wait_tensorcnt(1);      // slab blk landed
            } else {
                __builtin_amdgcn_s_wait_tensorcnt(0);
            }
        }
        __syncthreads();
        if (tid < 32) {   // 32 u64 words per 8-row slab
            int lr = tid >> 2, wc = tid & 3;
            const float* rp = &stage[blk & 1][lr * W_IMG + wc * 64];
            uint64_t bits = 0;
            for (int j = 0; j < 64; ++j)
                bits |= (rp[j] != 0.0f) ? (1ull << j) : 0ull;
            Bb[0][(blk * 8 + lr) * W64PR + wc] = bits;
        }
        __syncthreads();  // stage[blk&1] free for reuse by slab blk+2
    }

    if (tid == 0) chg[0] = 0;
    __syncthreads();

    // ------------- wavefront iterations, one u64 word per thread ------------
    const int r  = tid >> 2;                  // row 0..255
    const int wc = tid & 3;                   // u64 word-in-row 0..3
    const int rm = (r == 0) ? 0 : r - 1;      // replicate-pad row clamps
    const int rp = (r == H_IMG - 1) ? (H_IMG - 1) : r + 1;

    for (int it = 0; it < N_ITERS; ++it) {
        const int p = it & 1;
        if (tid == 0) chg[p ^ 1] = 0;         // prepare next iteration's flag

        const uint64_t* cur = Bb[p];
        const uint64_t* rowc = cur + r  * W64PR;
        const uint64_t* rown = cur + rm * W64PR;
        const uint64_t* rows = cur + rp * W64PR;

        uint64_t x  = rowc[wc];
        uint64_t xl = (wc > 0)         ? rowc[wc - 1] : 0ull;
        uint64_t xr = (wc < W64PR - 1) ? rowc[wc + 1] : 0ull;
        uint64_t nn = rown[wc];
        uint64_t nl = (wc > 0)         ? rown[wc - 1] : 0ull;
        uint64_t nr = (wc < W64PR - 1) ? rown[wc + 1] : 0ull;
        uint64_t ss = rows[wc];
        uint64_t sl = (wc > 0)         ? rows[wc - 1] : 0ull;
        uint64_t sr = (wc < W64PR - 1) ? rows[wc + 1] : 0ull;

        uint64_t aW = west64(x,  xl, wc), aE = east64(x,  xr, wc);
        uint64_t NW = west64(nn, nl, wc), NE = east64(nn, nr, wc);
        uint64_t SW = west64(ss, sl, wc), SE = east64(ss, sr, wc);

        uint64_t e0, e1, e2, d0, d1, d2;
        popplanes4(nn, ss, aW, aE, e0, e1, e2);   // edge-count bit planes
        popplanes4(NW, NE, SW, SE, d0, d1, d2);   // diag-count bit planes

        uint64_t any   = nn | ss | aW | aE | NW | NE | SW | SE;
        uint64_t newly = any & ~x;
        Bb[p ^ 1][tid] = x | newly;
        if (newly) chg[p] = 1;                    // benign same-value race

        const float off = (float)it;
        uint64_t m = newly;
        while (m) {   // each pixel enters here at most once over the whole run
            int j = __builtin_ctzll(m);
            m &= m - 1;
            int ne = (int)((e0 >> j) & 1ull) | ((int)((e1 >> j) & 1ull) << 1)
                   | ((int)((e2 >> j) & 1ull) << 2);
            int nc = (int)((d0 >> j) & 1ull) | ((int)((d1 >> j) & 1ull) << 1)
                   | ((int)((d2 >> j) & 1ull) << 2);
            ob[r * W_IMG + wc * 64 + j] = off + lut[ne * 5 + nc];
        }

        __syncthreads();
        if (chg[p] == 0) break;   // wavefront saturated: remaining iterations
                                  // are exact no-ops in the reference too
    }
}

extern "C" void kernel_launch(void* const* d_in, const int* in_sizes, int n_in,
                              void* d_out, int out_size, void* d_ws, size_t ws_size,
                              hipStream_t stream) {
    (void)n_in; (void)out_size; (void)d_ws; (void)ws_size;
    const float* img = (const float*)d_in[0];
    float*       out = (float*)d_out;
    int batch = in_sizes[0] / (H_IMG * W_IMG);
    if (batch <= 0) batch = 1;
    cdt_wavefront_kernel<<<dim3(batch), dim3(NTHREADS), 0, stream>>>(img, out);
}